// BatchRankingLoss_2748779070189
// MI455X (gfx1250) — compile-verified
//
#include <hip/hip_runtime.h>
#include <hip/hip_bf16.h>

// ---------------------------------------------------------------------------
// BatchRankingLoss on MI455X (gfx1250, wave32).
//   loss = (1/N) * sum_{i,j} w_ij * max(0, 1 + y_ij*(o_i - o_j))
//   y_ij = (l_i < l_j) ? -1 : +1 ;  w_ij = |l_i - l_j| > 0.1
// The relu makes the O(B^2) part non-factorizable -> it runs on the VALU pipe
// at ~7 ops/pair from LDS-resident tiles. The f32 matrix pipe
// (V_WMMA_F32_16X16X4_F32, ones-B operand) performs the cross-lane reductions
// at full f32 precision. Inner loop reads interleaved (l,o) pairs as float4
// so one ds_load_b128 feeds two pairs and waits amortize over 8 pairs.
// ---------------------------------------------------------------------------

typedef __attribute__((ext_vector_type(2))) float v2f;
typedef __attribute__((ext_vector_type(8))) float v8f;

#define TPB 256      // 8 wave32s per block
#define JT  512      // j-tile staged in LDS (512 * float2 = 4 KB)

// Full-wave (32 lane) f32 sum on the matrix pipe.
// A (16x4 f32, 2 VGPRs): VGPR0 lanes0-15 -> A[m][0]=acc[m], lanes16-31 ->
// A[m][2]=acc[m+16]; VGPR1 = 0. B = ones => D[m][n] = acc[m]+acc[m+16].
// Lane L<16 holds rows 0..7 in d[0..7]; lane L>=16 holds rows 8..15.
// Sum d[0..7] in-lane, then xor-shuffle across the half-wave boundary.
// EXEC must be all ones at the call site (all callers guarantee this).
__device__ __forceinline__ float wave_sum_wmma(float acc) {
    v2f a; a[0] = acc;  a[1] = 0.0f;
    v2f b; b[0] = 1.0f; b[1] = 1.0f;
    v8f c = {};
    v8f d = __builtin_amdgcn_wmma_f32_16x16x4_f32(
        /*neg_a=*/false, a, /*neg_b=*/false, b,
        /*c_mod=*/(short)0, c, /*reuse_a=*/false, /*reuse_b=*/false);
    float s = 0.0f;
#pragma unroll
    for (int k = 0; k < 8; ++k) s += d[k];
    s += __shfl_xor(s, 16, 32);   // rows 0..7 + rows 8..15 -> full wave sum
    return s;                     // every lane holds the wave total
}

// Weighted hinge for one (i,j) pair.
__device__ __forceinline__ float pair_term(float li, float oi, float lj, float oj) {
    const float y = (li < lj) ? -1.0f : 1.0f;
    const float h = fmaxf(0.0f, fmaf(y, oi - oj, 1.0f));
    return (fabsf(li - lj) > 0.1f) ? h : 0.0f;   // diagonal weight = 0
}

__global__ void __launch_bounds__(TPB)
pairwise_hinge_kernel(const float* __restrict__ o,   // [B] (input squeezed)
                      const float* __restrict__ l,   // [B] gdt_ts
                      float* __restrict__ partials,  // [gridDim.x*gridDim.y]
                      int B) {
    __shared__ __align__(16) float2 s_lo[JT];        // interleaved (l, o)
    __shared__ float s_wave[TPB / 32];

    const int tid = threadIdx.x;
    const int i   = blockIdx.x * TPB + tid;
    const int j0  = blockIdx.y * JT;
    const int jt  = min(JT, B - j0);

    // Stage the j-tile into LDS (coalesced, shared by all 8 waves).
    for (int t = tid; t < jt; t += TPB)
        s_lo[t] = make_float2(l[j0 + t], o[j0 + t]);
    __syncthreads();

    float acc = 0.0f;
    if (i < B) {
        const float oi = o[i];
        const float li = l[i];
        const float4* __restrict__ p4 = reinterpret_cast<const float4*>(s_lo);
        const int n4 = jt >> 1;                      // float4 = two (l,o) pairs
#pragma unroll 4
        for (int q = 0; q < n4; ++q) {
            const float4 v = p4[q];                  // ds_load_b128: 2 pairs
            acc += pair_term(li, oi, v.x, v.y);
            acc += pair_term(li, oi, v.z, v.w);
        }
        if (jt & 1) {
            const float2 v = s_lo[jt - 1];
            acc += pair_term(li, oi, v.x, v.y);
        }
    }

    // Wave reduction on the matrix pipe (all 256 threads active here).
    const float ws = wave_sum_wmma(acc);
    if ((tid & 31) == 0) s_wave[tid >> 5] = ws;
    __syncthreads();

    if (tid == 0) {
        float t = 0.0f;
#pragma unroll
        for (int k = 0; k < TPB / 32; ++k) t += s_wave[k];
        partials[blockIdx.y * gridDim.x + blockIdx.x] = t;
    }
}

__global__ void __launch_bounds__(TPB)
final_reduce_kernel(const float* __restrict__ partials, int P,
                    float* __restrict__ out, float invN) {
    __shared__ float s_wave[TPB / 32];
    const int tid = threadIdx.x;

    float acc = 0.0f;
    for (int t = tid; t < P; t += TPB) acc += partials[t];

    const float ws = wave_sum_wmma(acc);   // all 256 threads active
    if ((tid & 31) == 0) s_wave[tid >> 5] = ws;
    __syncthreads();

    if (tid == 0) {
        float t = 0.0f;
#pragma unroll
        for (int k = 0; k < TPB / 32; ++k) t += s_wave[k];
        out[0] = t * invN;
    }
}

extern "C" void kernel_launch(void* const* d_in, const int* in_sizes, int n_in,
                              void* d_out, int out_size, void* d_ws, size_t ws_size,
                              hipStream_t stream) {
    const float* o = (const float*)d_in[0];   // input  [B,1] -> flat [B]
    const float* l = (const float*)d_in[1];   // gdt_ts [B]
    float* out      = (float*)d_out;
    float* partials = (float*)d_ws;

    const int B  = in_sizes[1];               // 8192
    const int gx = (B + TPB - 1) / TPB;       // 32 i-blocks
    const int gy = (B + JT  - 1) / JT;        // 16 j-blocks
    const int P  = gx * gy;                   // 512 block partials (2 KB in d_ws)

    const double N    = (double)B * (double)(B - 1);
    const float  invN = (float)(1.0 / N);

    pairwise_hinge_kernel<<<dim3(gx, gy), TPB, 0, stream>>>(o, l, partials, B);
    final_reduce_kernel<<<1, TPB, 0, stream>>>(partials, P, out, invN);
}